// SoftSplat_49830210568300
// MI455X (gfx1250) — compile-verified
//
#include <hip/hip_runtime.h>
#include <cstdint>

// Problem constants (from reference setup_inputs)
#define BATCH 4
#define CH    64
#define HH    256
#define WW    448
#define HW    (HH * WW)        // 114688
#define CHW   (CH * HW)        // 7340032
#define NPIX  (BATCH * HW)     // 458752
#define NOUT  (BATCH * CHW)    // 29360128
#define CH_PER_Y 16            // channels handled per gridDim.y slice

// Native f32 global atomic add (GLOBAL_ATOMIC_ADD_F32, no-return -> STOREcnt).
__device__ __forceinline__ void gatomic_add(float* p, float v) {
    unsafeAtomicAdd(p, v);
}

// blockDim.x = WW (448 = 14 wave32s), grid = (HH, CH/CH_PER_Y, BATCH).
// All indices come from grid geometry -> zero integer divisions.
__global__ __launch_bounds__(WW) void splat_kernel(const float* __restrict__ inp,
                                                   const float* __restrict__ flow,
                                                   const float* __restrict__ mask,
                                                   float* __restrict__ out,
                                                   float* __restrict__ wacc) {
    const int x  = threadIdx.x;
    const int y  = blockIdx.x;
    const int b  = blockIdx.z;
    const int c0 = blockIdx.y * CH_PER_Y;
    const int pix = y * WW + x;

    // flow layout [B,2,H,W]: channel 0 = dx, channel 1 = dy
    const float dx = flow[(size_t)(b * 2 + 0) * HW + pix];
    const float dy = flow[(size_t)(b * 2 + 1) * HW + pix];
    const float m  = __expf(mask[(size_t)b * HW + pix]);   // v_exp_f32

    const float tx  = (float)x + dx;
    const float ty  = (float)y + dy;
    const float x0f = floorf(tx);
    const float y0f = floorf(ty);
    const int   x0  = (int)x0f;
    const int   y0  = (int)y0f;
    const float wx1 = tx - x0f, wx0 = 1.0f - wx1;
    const float wy1 = ty - y0f, wy0 = 1.0f - wy1;

    // 4 bilinear corners (NW, NE, SW, SE); OOB contributions dropped
    const float cw[4] = { wx0 * wy0, wx1 * wy0, wx0 * wy1, wx1 * wy1 };
    const int   cx[4] = { x0, x0 + 1, x0,     x0 + 1 };
    const int   cy[4] = { y0, y0,     y0 + 1, y0 + 1 };
    bool valid[4];
    int  cidx[4];
#pragma unroll
    for (int k = 0; k < 4; ++k) {
        valid[k] = (cx[k] >= 0) & (cx[k] < WW) & (cy[k] >= 0) & (cy[k] < HH);
        cidx[k]  = cy[k] * WW + cx[k];
    }

    const float* ip = inp + (size_t)b * CHW + (size_t)c0 * HW + pix;
    float*       op = out + (size_t)b * CHW + (size_t)c0 * HW;

    // Phase 1: issue all 16 channel loads (coalesced; 16 outstanding loads)
    float v[CH_PER_Y];
#pragma unroll
    for (int c = 0; c < CH_PER_Y; ++c)
        v[c] = ip[(size_t)c * HW];

    // Phase 2: 64 no-return f32 atomics into the L2-resident accumulator
#pragma unroll
    for (int c = 0; c < CH_PER_Y; ++c) {
        const float val = v[c] * m;
        float* ob = op + (size_t)c * HW;
#pragma unroll
        for (int k = 0; k < 4; ++k)
            if (valid[k]) gatomic_add(ob + cidx[k], val * cw[k]);
    }

    // chunk 0 also splats the weight (m) channel into the workspace plane
    if (blockIdx.y == 0) {
        float* wb = wacc + (size_t)b * HW;
#pragma unroll
        for (int k = 0; k < 4; ++k)
            if (valid[k]) gatomic_add(wb + cidx[k], m * cw[k]);
    }
}

__global__ __launch_bounds__(256) void zero_kernel(float4* __restrict__ out,
                                                   float4* __restrict__ wacc,
                                                   int nout4, int nw4) {
    int idx    = blockIdx.x * 256 + threadIdx.x;
    int stride = gridDim.x * 256;
    const float4 z = { 0.f, 0.f, 0.f, 0.f };
    for (int i = idx; i < nout4; i += stride) out[i] = z;   // global_store_b128
    for (int i = idx; i < nw4;   i += stride) wacc[i] = z;
}

// grid = (HW/4/256, CH, BATCH): one float4 per thread, no divisions.
__global__ __launch_bounds__(256) void norm_kernel(float* __restrict__ out,
                                                   const float* __restrict__ wacc) {
    const int pix4 = (blockIdx.x * 256 + threadIdx.x) * 4;  // 16B aligned
    const int c    = blockIdx.y;
    const int b    = blockIdx.z;
    float4 wv = *(const float4*)(wacc + (size_t)b * HW + pix4);           // b128
    float4 o  = *(float4*)(out + (size_t)(b * CH + c) * HW + pix4);       // b128
    o.x = o.x / (wv.x + 1e-7f);
    o.y = o.y / (wv.y + 1e-7f);
    o.z = o.z / (wv.z + 1e-7f);
    o.w = o.w / (wv.w + 1e-7f);
    *(float4*)(out + (size_t)(b * CH + c) * HW + pix4) = o;               // b128
}

extern "C" void kernel_launch(void* const* d_in, const int* in_sizes, int n_in,
                              void* d_out, int out_size, void* d_ws, size_t ws_size,
                              hipStream_t stream) {
    const float* inp  = (const float*)d_in[0];   // [4,64,256,448] f32
    const float* flow = (const float*)d_in[1];   // [4,2,256,448]  f32
    const float* mask = (const float*)d_in[2];   // [4,1,256,448]  f32
    float* out  = (float*)d_out;                 // [4,64,256,448] f32
    float* wacc = (float*)d_ws;                  // [4,256,448]    f32 weight accumulator

    // 1) zero accumulators (harness poisons buffers; do this every call)
    zero_kernel<<<2048, 256, 0, stream>>>((float4*)out, (float4*)wacc,
                                          NOUT / 4, NPIX / 4);

    // 2) forward splat: one 448-thread block per (row, 16-channel chunk, batch)
    dim3 sgrid(HH, CH / CH_PER_Y, BATCH);
    splat_kernel<<<sgrid, WW, 0, stream>>>(inp, flow, mask, out, wacc);

    // 3) normalize by splatted weight channel (float4 per thread)
    dim3 ngrid(HW / 4 / 256, CH, BATCH);
    norm_kernel<<<ngrid, 256, 0, stream>>>(out, wacc);

    (void)in_sizes; (void)n_in; (void)out_size; (void)ws_size;
}